// SingleQueryAttention_40785009443303
// MI455X (gfx1250) — compile-verified
//
#include <hip/hip_runtime.h>
#include <math.h>

// ---------------------------------------------------------------------------
// Single-query attention for MI455X (gfx1250, wave32).
// D_X = D_Z = D_ATTN = D_OUT = 2048, L_Z = 32768.
// Outputs (flat, in order): q[2048], k[L*2048], v[L*2048], score[L], mask[L],
//                           masked_score[L], attention[L], vt[2048]
// ---------------------------------------------------------------------------

#define D    2048
#define LZ   32768

typedef __attribute__((ext_vector_type(2))) float v2f;
typedef __attribute__((ext_vector_type(8))) float v8f;
typedef __attribute__((__vector_size__(4 * sizeof(int)))) int i4;

// Async global->LDS path (CDNA5): gated on builtin availability so the file
// compiles on any toolchain; fallback is the known-good synchronous staging.
#if __has_builtin(__builtin_amdgcn_global_load_async_to_lds_b128) && \
    __has_builtin(__builtin_amdgcn_global_load_async_to_lds_b32)  && \
    __has_builtin(__builtin_amdgcn_s_wait_asynccnt)
#define HAVE_ASYNC 1
#else
#define HAVE_ASYNC 0
#endif

// Typed, address-space-qualified casts matching the builtin signatures:
//   b128: (i4 as1*, i4 as3*, Imm int, Imm int)
//   b32 : (int as1*, int as3*, Imm int, Imm int)
#define GPTR128(p) ((__attribute__((address_space(1))) i4*)(void*)(p))
#define SPTR128(p) ((__attribute__((address_space(3))) i4*)(void*)(p))
#define GPTR32(p)  ((__attribute__((address_space(1))) int*)(void*)(p))
#define SPTR32(p)  ((__attribute__((address_space(3))) int*)(void*)(p))

// ---------------------------------------------------------------------------
// q = x @ w_q (partial over i-chunks, accumulated with atomics into d_ws)
// ---------------------------------------------------------------------------
__global__ __launch_bounds__(256) void q_partial_kernel(
    const float* __restrict__ x, const float* __restrict__ wq,
    float* __restrict__ qacc) {
  __shared__ float xl[128];
  const int tid = threadIdx.x;
  const int j   = blockIdx.x * 256 + tid;
  const int i0  = blockIdx.y * 128;
  if (tid < 128) xl[tid] = x[i0 + tid];
  __syncthreads();
  float acc = 0.0f;
#pragma unroll 8
  for (int i = 0; i < 128; ++i)
    acc += xl[i] * wq[(size_t)(i0 + i) * D + j];
  atomicAdd(&qacc[j], acc);
}

__global__ __launch_bounds__(256) void q_final_kernel(
    const float* __restrict__ qacc, const float* __restrict__ bq,
    float* __restrict__ qout) {
  const int j = blockIdx.x * 256 + threadIdx.x;
  qout[j] = qacc[j] + bq[j];
}

// ---------------------------------------------------------------------------
// K = zs @ w_k + b_k  and  V = zs @ w_v + b_v   (blockIdx.z selects which)
// Block tile 64x32, K-step 32, 8 waves each owning one 16x16 WMMA C tile.
// V_WMMA_F32_16X16X4_F32 (full f32 precision; workload is HBM-bound).
// Async double-buffered LDS staging when the toolchain exposes
// GLOBAL_LOAD_ASYNC_TO_LDS (ASYNCcnt pipeline), else synchronous staging.
// ---------------------------------------------------------------------------
#define BM 64
#define BN 32
#define BK 32
#define LDA 36   // stride 144B: 16B-aligned b128 stores, conflict-free b64 frag reads
#define LDB 36

#if HAVE_ASYNC
#define NBUF 2
#else
#define NBUF 1
#endif

__global__ __launch_bounds__(256) void kv_gemm_kernel(
    const float* __restrict__ zs,
    const float* __restrict__ wk, const float* __restrict__ wv,
    const float* __restrict__ bk, const float* __restrict__ bv,
    float* __restrict__ kout, float* __restrict__ vout) {
  __shared__ __align__(16) float Alds[NBUF][BM][LDA];  // zs tile [m][k]
  __shared__ __align__(16) float Blds[NBUF][BN][LDB];  // w tile transposed [n][k]

  const int which = blockIdx.z;
  const float* __restrict__ w    = which ? wv : wk;
  const float* __restrict__ bias = which ? bv : bk;
  float* __restrict__ out        = which ? vout : kout;

  const int n_base = blockIdx.x * BN;   // x fastest -> A-stripe reuse in L2
  const int m_base = blockIdx.y * BM;

  const int tid  = threadIdx.x;
  const int lane = tid & 31;
  const int wave = tid >> 5;         // 0..7
  const int m0   = (wave & 3) * 16;
  const int n0   = (wave >> 2) * 16;
  const int lm   = lane & 15;
  const int hi   = lane >> 4;        // 0: K pair {0,1}   1: K pair {2,3}

  // tile-load mappings
  const int ar  = tid >> 3;          // A row (0..31; second slot +32)
  const int ac4 = (tid & 7) * 4;     // A col group
  const int bn  = tid & 31;          // B col (n)
  const int bkr = tid >> 5;          // B row base (k), +8 per j

  v8f c = {0.f,0.f,0.f,0.f,0.f,0.f,0.f,0.f};

#if HAVE_ASYNC
  auto issue_tile = [&](int buf, int k0) {
    // A: 2 x b128 per thread (64x32 floats)
    __builtin_amdgcn_global_load_async_to_lds_b128(
        GPTR128(&zs[(size_t)(m_base + ar) * D + k0 + ac4]),
        SPTR128(&Alds[buf][ar][ac4]), 0, 0);
    __builtin_amdgcn_global_load_async_to_lds_b128(
        GPTR128(&zs[(size_t)(m_base + ar + 32) * D + k0 + ac4]),
        SPTR128(&Alds[buf][ar + 32][ac4]), 0, 0);
    // B: 4 x b32 per thread, transposed into [n][k]
#pragma unroll
    for (int j = 0; j < 4; ++j) {
      const int kk = bkr + 8 * j;
      __builtin_amdgcn_global_load_async_to_lds_b32(
          GPTR32(&w[(size_t)(k0 + kk) * D + n_base + bn]),
          SPTR32(&Blds[buf][bn][kk]), 0, 0);
    }
  };

  issue_tile(0, 0);
  for (int k0 = 0; k0 < D; k0 += BK) {
    const int cur = (k0 >> 5) & 1;
    if (k0 + BK < D) {
      issue_tile(cur ^ 1, k0 + BK);              // prefetch next tile
      __builtin_amdgcn_s_wait_asynccnt(6);       // wait only for CURRENT tile
    } else {
      __builtin_amdgcn_s_wait_asynccnt(0);       // last tile: drain
    }
    __syncthreads();                             // tile visible to all waves
#pragma unroll
    for (int kk = 0; kk < BK; kk += 4) {
      v2f a = *(const v2f*)&Alds[cur][m0 + lm][kk + 2 * hi];
      v2f b = *(const v2f*)&Blds[cur][n0 + lm][kk + 2 * hi];
      c = __builtin_amdgcn_wmma_f32_16x16x4_f32(false, a, false, b,
                                                (short)0, c, false, false);
    }
    __syncthreads();                             // safe to overwrite buffer
  }
#else
  for (int k0 = 0; k0 < D; k0 += BK) {
    __syncthreads();
    {
      const float4 a0 = *(const float4*)&zs[(size_t)(m_base + ar) * D + k0 + ac4];
      const float4 a1 = *(const float4*)&zs[(size_t)(m_base + ar + 32) * D + k0 + ac4];
      *(float4*)&Alds[0][ar][ac4]      = a0;     // 16B aligned (stride 144B)
      *(float4*)&Alds[0][ar + 32][ac4] = a1;
    }
#pragma unroll
    for (int j = 0; j < 4; ++j) {
      const int kk = bkr + 8 * j;
      Blds[0][bn][kk] = w[(size_t)(k0 + kk) * D + n_base + bn];
    }
    __syncthreads();
#pragma unroll
    for (int kk = 0; kk < BK; kk += 4) {
      v2f a = *(const v2f*)&Alds[0][m0 + lm][kk + 2 * hi];
      v2f b = *(const v2f*)&Blds[0][n0 + lm][kk + 2 * hi];
      c = __builtin_amdgcn_wmma_f32_16x16x4_f32(false, a, false, b,
                                                (short)0, c, false, false);
    }
  }
#endif

  // Epilogue: C layout — VGPR r: lanes 0-15 = row r, lanes 16-31 = row r+8
  const int col = n_base + n0 + lm;
  const float biasv = bias[col];
#pragma unroll
  for (int r = 0; r < 8; ++r) {
    const int row = m_base + m0 + r + 8 * hi;
    out[(size_t)row * D + col] = c[r] + biasv;
  }
}

// ---------------------------------------------------------------------------
// score[l] = dot(K[l], q) * scale     one wave per row
// ---------------------------------------------------------------------------
__global__ __launch_bounds__(256) void score_kernel(
    const float* __restrict__ K, const float* __restrict__ q,
    float* __restrict__ score, float scale) {
  const int row  = blockIdx.x * 8 + (threadIdx.x >> 5);
  const int lane = threadIdx.x & 31;
  const float* kr = K + (size_t)row * D;
  float acc = 0.0f;
#pragma unroll 4
  for (int i = lane * 4; i < D; i += 32 * 4) {
    const float4 kv = *(const float4*)(kr + i);
    const float4 qv = *(const float4*)(q + i);
    acc += kv.x * qv.x + kv.y * qv.y + kv.z * qv.z + kv.w * qv.w;
  }
#pragma unroll
  for (int off = 16; off > 0; off >>= 1) acc += __shfl_xor(acc, off, 32);
  if (lane == 0) score[row] = acc * scale;
}

// ---------------------------------------------------------------------------
// mask / masked_score / softmax / attention — single block, 1024 threads
// ---------------------------------------------------------------------------
__global__ __launch_bounds__(1024) void softmax_kernel(
    const float* __restrict__ score, const int* __restrict__ zs_mask,
    const int* __restrict__ x_mask, float* __restrict__ maskf,
    float* __restrict__ masked, float* __restrict__ attn) {
  __shared__ float red[1024];
  const int tid = threadIdx.x;
  const int xm  = x_mask[0];
  const float NEG = -3.402823466e38f;  // jnp.finfo(float32).min

  float lmax = NEG;
  for (int l = tid; l < LZ; l += 1024) {
    const int   m  = xm * zs_mask[l];
    const float s  = score[l];
    const float ms = (m != 0) ? s : NEG;
    maskf[l]  = (float)m;
    masked[l] = ms;
    lmax = fmaxf(lmax, ms);
  }
  red[tid] = lmax;
  __syncthreads();
  for (int s = 512; s > 0; s >>= 1) {
    if (tid < s) red[tid] = fmaxf(red[tid], red[tid + s]);
    __syncthreads();
  }
  const float gmax = red[0];
  __syncthreads();

  float lsum = 0.0f;
  for (int l = tid; l < LZ; l += 1024) {
    const float e = expf(masked[l] - gmax);
    attn[l] = e;           // unnormalized
    lsum += e;
  }
  red[tid] = lsum;
  __syncthreads();
  for (int s = 512; s > 0; s >>= 1) {
    if (tid < s) red[tid] += red[tid + s];
    __syncthreads();
  }
  const float inv = 1.0f / red[0];
  __syncthreads();

  for (int l = tid; l < LZ; l += 1024)
    attn[l] = attn[l] * inv * maskf[l];
}

// ---------------------------------------------------------------------------
// vt = attention @ V : two-stage column reduction.
// ---------------------------------------------------------------------------
__global__ __launch_bounds__(256) void vt_partial_kernel(
    const float* __restrict__ V, const float* __restrict__ attn,
    float* __restrict__ partial) {
  __shared__ float alds[2048];
  const int tid = threadIdx.x;
  const int col = blockIdx.x * 256 + tid;
  const int l0  = blockIdx.y * 2048;
  for (int i = tid; i < 2048; i += 256) alds[i] = attn[l0 + i];
  __syncthreads();
  float acc = 0.0f;
#pragma unroll 8
  for (int i = 0; i < 2048; ++i)
    acc += alds[i] * V[(size_t)(l0 + i) * D + col];
  partial[(size_t)blockIdx.y * D + col] = acc;
}

__global__ __launch_bounds__(256) void vt_final_kernel(
    const float* __restrict__ partial, float* __restrict__ vt) {
  const int col = blockIdx.x * 256 + threadIdx.x;
  float acc = 0.0f;
#pragma unroll
  for (int p = 0; p < 16; ++p) acc += partial[(size_t)p * D + col];
  vt[col] = acc;
}

// ---------------------------------------------------------------------------
extern "C" void kernel_launch(void* const* d_in, const int* in_sizes, int n_in,
                              void* d_out, int out_size, void* d_ws, size_t ws_size,
                              hipStream_t stream) {
  (void)in_sizes; (void)n_in; (void)out_size; (void)ws_size;

  const float* x       = (const float*)d_in[0];
  const float* zs      = (const float*)d_in[1];
  const int*   x_mask  = (const int*)d_in[2];
  const int*   zs_mask = (const int*)d_in[3];
  const float* w_q     = (const float*)d_in[4];
  const float* w_k     = (const float*)d_in[5];
  const float* w_v     = (const float*)d_in[6];
  const float* b_q     = (const float*)d_in[7];
  const float* b_k     = (const float*)d_in[8];
  const float* b_v     = (const float*)d_in[9];

  float* out    = (float*)d_out;
  float* q_out  = out;
  float* k_out  = q_out + D;
  float* v_out  = k_out + (size_t)LZ * D;
  float* score  = v_out + (size_t)LZ * D;
  float* maskf  = score + LZ;
  float* masked = maskf + LZ;
  float* attn   = masked + LZ;
  float* vt     = attn + LZ;

  float* qacc    = (float*)d_ws;       // 2048 floats
  float* partial = qacc + D;           // 16 * 2048 floats

  const float scale = 1.0f / sqrtf((float)D);

  // q = x @ w_q + b_q
  (void)hipMemsetAsync(qacc, 0, D * sizeof(float), stream);
  q_partial_kernel<<<dim3(8, 16), 256, 0, stream>>>(x, w_q, qacc);
  q_final_kernel<<<8, 256, 0, stream>>>(qacc, b_q, q_out);

  // K, V GEMMs (WMMA f32, async double-buffered staging)
  kv_gemm_kernel<<<dim3(D / BN, LZ / BM, 2), 256, 0, stream>>>(
      zs, w_k, w_v, b_k, b_v, k_out, v_out);

  // score = (K @ q) * scale
  score_kernel<<<LZ / 8, 256, 0, stream>>>(k_out, q_out, score, scale);

  // mask / masked_score / attention
  softmax_kernel<<<1, 1024, 0, stream>>>(score, zs_mask, x_mask, maskf, masked, attn);

  // vt = attention @ V
  vt_partial_kernel<<<dim3(8, 16), 256, 0, stream>>>(v_out, attn, partial);
  vt_final_kernel<<<8, 256, 0, stream>>>(partial, vt);
}